// ThresholdNMS_13683765805629
// MI455X (gfx1250) — compile-verified
//
#include <hip/hip_runtime.h>
#include <stdint.h>

#define N_BOXES_C   100800
#define N_FEATS_C   85
#define MAX_DET_C   100
#define IOU_THR_C   0.45f
#define NMS_WG_C    96
#define NMS_BLK_C   256
#define NTILES_C    (N_BOXES_C / 16)   // 6300 tiles of 16 boxes
#define NEG_INF_F   (-__builtin_inff())

typedef __attribute__((ext_vector_type(2))) float v2f;
typedef __attribute__((ext_vector_type(8))) float v8f;

__device__ __forceinline__ unsigned long long make_key(float s, unsigned int i) {
    // scores are >= 0 (products of uniforms), so float bits are monotone.
    // ~i in the low bits => for equal scores, smaller index wins the max
    // (replicates stable argsort(-scores) tie-breaking).
    return ((unsigned long long)__float_as_uint(s) << 32) |
           (unsigned long long)(~i);
}

// ---------------------------------------------------------------------------
// Init: zero the per-pick argmax key slots and the grid-barrier state.
// (d_ws is poisoned / left dirty between replays, so this runs every launch.)
// ---------------------------------------------------------------------------
__global__ void nms_init_kernel(unsigned long long* __restrict__ keys,
                                unsigned int* __restrict__ bar) {
    int t = threadIdx.x;
    if (t < MAX_DET_C + 1) keys[t] = 0ull;
    if (t < 2) bar[t] = 0u;
}

// ---------------------------------------------------------------------------
// Box transform: boxes = preds[:,0:4] @ convert, using V_WMMA_F32_16X16X4_F32.
// One wave handles 16 boxes (rows). ISA VGPR layouts (CDNA5 ISA):
//   A (16x4 f32):  lanes 0-15: M=lane,    vgpr0=K0, vgpr1=K1
//                  lanes16-31: M=lane-16, vgpr0=K2, vgpr1=K3
//   B (4x16 f32):  vgpr0: lanes0-15 row K0, lanes16-31 row K2
//                  vgpr1: lanes0-15 row K1, lanes16-31 row K3
//   D (16x16 f32): vgpr v: lanes0-15 (M=v, N=lane); lanes16-31 (M=v+8, N=lane-16)
// convert rows: r0=[1,0,1,0], r1=[0,1,0,1], r2=[-.5,0,.5,0], r3=[0,-.5,0,.5]
// ---------------------------------------------------------------------------
__global__ void box_wmma_kernel(const float* __restrict__ preds,
                                float* __restrict__ boxes) {
    int wave = blockIdx.x * (blockDim.x >> 5) + (threadIdx.x >> 5);
    if (wave >= NTILES_C) return;          // wave-uniform; EXEC all-1 for WMMA
    int lane = threadIdx.x & 31;
    int base = wave << 4;
    int row  = base + (lane & 15);
    int k0   = (lane >> 4) << 1;           // lanes 0-15 -> K0,K1 ; 16-31 -> K2,K3
    const float* p = preds + (size_t)row * N_FEATS_C + k0;
    v2f a;
    a.x = p[0];
    a.y = p[1];

    float b0 = 0.f, b1 = 0.f;              // B matrix (convert), per-lane
    if (lane == 0 || lane == 2)  b0 = 1.0f;     // row0: cols 0,2 = 1
    if (lane == 16)              b0 = -0.5f;    // row2: col 0 = -0.5
    if (lane == 18)              b0 =  0.5f;    // row2: col 2 =  0.5
    if (lane == 1 || lane == 3)  b1 = 1.0f;     // row1: cols 1,3 = 1
    if (lane == 17)              b1 = -0.5f;    // row3: col 1 = -0.5
    if (lane == 19)              b1 =  0.5f;    // row3: col 3 =  0.5
    v2f b;
    b.x = b0;
    b.y = b1;
    v8f c = {};

#if __has_builtin(__builtin_amdgcn_wmma_f32_16x16x4_f32)
    v8f d = __builtin_amdgcn_wmma_f32_16x16x4_f32(
        /*neg_a=*/false, a, /*neg_b=*/false, b,
        /*c_mod=*/(short)0, c, /*reuse_a=*/false, /*reuse_b=*/false);
#else
    // Scalar fallback matching the D layout (should not trigger on gfx1250).
    v8f d = c;
    {
        int colf  = lane & 15;
        int rofsf = (lane >> 4) * 8;
#pragma unroll
        for (int v = 0; v < 8; ++v) {
            int r = base + rofsf + v;
            const float* q = preds + (size_t)r * N_FEATS_C;
            float x = q[0], y = q[1], wd = q[2], h = q[3];
            float val = 0.f;
            if (colf == 0) val = x - 0.5f * wd;
            else if (colf == 1) val = y - 0.5f * h;
            else if (colf == 2) val = x + 0.5f * wd;
            else if (colf == 3) val = y + 0.5f * h;
            d[v] = val;
        }
    }
#endif

    int col  = lane & 15;
    int rofs = (lane >> 4) * 8;
    if (col < 4) {
#pragma unroll
        for (int v = 0; v < 8; ++v) {
            boxes[(size_t)(base + rofs + v) * 4 + col] = d[v];
        }
    }
}

// ---------------------------------------------------------------------------
// Scores: scores[i] = preds[i][4] * max(preds[i][5:85]).
// One wave per box (coalesced 32-lane reads of the class scores), butterfly
// max reduction, then hierarchical (wave -> block -> atomicMax) seeding of
// keys[0] = argmax key over all boxes for the first NMS pick.
// ---------------------------------------------------------------------------
__global__ void score_kernel(const float* __restrict__ preds,
                             float* __restrict__ scores,
                             unsigned long long* __restrict__ keys) {
    const int wavesPerBlk = blockDim.x >> 5;
    const int wave = blockIdx.x * wavesPerBlk + (threadIdx.x >> 5);
    const int lane = threadIdx.x & 31;
    const int nw   = gridDim.x * wavesPerBlk;

    unsigned long long best = 0ull;
    for (int box = wave; box < N_BOXES_C; box += nw) {
        const float* row = preds + (size_t)box * N_FEATS_C;
        if (box + nw < N_BOXES_C)  // emits global_prefetch_b8
            __builtin_prefetch(preds + (size_t)(box + nw) * N_FEATS_C + lane * 8, 0, 0);
        float m = row[5 + lane];                 // classes 5..36
        m = fmaxf(m, row[37 + lane]);            // classes 37..68
        if (lane < 16) m = fmaxf(m, row[69 + lane]);  // classes 69..84
#pragma unroll
        for (int off = 16; off >= 1; off >>= 1)
            m = fmaxf(m, __shfl_xor(m, off, 32));
        float s = row[4] * m;
        if (lane == 0) {
            scores[box] = s;
            unsigned long long k = make_key(s, (unsigned int)box);
            if (k > best) best = k;
        }
    }

    __shared__ unsigned long long sb[8];
    if (lane == 0) sb[threadIdx.x >> 5] = best;
    __syncthreads();
    if (threadIdx.x == 0) {
        unsigned long long b = sb[0];
        for (int w = 1; w < wavesPerBlk; ++w)
            if (sb[w] > b) b = sb[w];
        if (b != 0ull) atomicMax(keys, b);       // global_atomic_max_u64
    }
}

// ---------------------------------------------------------------------------
// Device-wide sense-reversing barrier (agent-scope atomics, s_sleep spin).
// All NMS_WG_C blocks are trivially co-resident on MI455X.
// ---------------------------------------------------------------------------
__device__ __forceinline__ void grid_barrier(unsigned int* count,
                                             unsigned int* gen,
                                             unsigned int nwg) {
    __syncthreads();
    if (threadIdx.x == 0) {
        unsigned int g = __hip_atomic_load(gen, __ATOMIC_ACQUIRE,
                                           __HIP_MEMORY_SCOPE_AGENT);
        unsigned int prev = __hip_atomic_fetch_add(count, 1u, __ATOMIC_ACQ_REL,
                                                   __HIP_MEMORY_SCOPE_AGENT);
        if (prev == nwg - 1u) {
            __hip_atomic_store(count, 0u, __ATOMIC_RELAXED,
                               __HIP_MEMORY_SCOPE_AGENT);
            __hip_atomic_fetch_add(gen, 1u, __ATOMIC_ACQ_REL,
                                   __HIP_MEMORY_SCOPE_AGENT);
        } else {
            while (__hip_atomic_load(gen, __ATOMIC_ACQUIRE,
                                     __HIP_MEMORY_SCOPE_AGENT) == g) {
                __builtin_amdgcn_s_sleep(2);
            }
        }
    }
    __syncthreads();
}

// ---------------------------------------------------------------------------
// Persistent exact greedy NMS. Iteration k:
//   * keys[k] already holds the max (score,~idx) key over unsuppressed boxes
//   * pick it; in ONE pass each thread suppresses its elements vs the pick
//     (scores[i] = -inf, thread-private partition => race-free) and folds the
//     survivors into a block max for keys[k+1]; one atomicMax per block.
//   * one grid barrier per pick (100 total device-wide syncs).
// Matches torchvision/reference semantics exactly, incl. keep[k] = -1 when
// every box is suppressed (key slot stays 0).
// ---------------------------------------------------------------------------
__global__ void nms_kernel(const float4* __restrict__ boxes,
                           float* __restrict__ scores,
                           unsigned long long* __restrict__ keys,
                           int* __restrict__ keep,
                           unsigned int* __restrict__ bar) {
    const int tid   = blockIdx.x * blockDim.x + threadIdx.x;
    const int total = gridDim.x * blockDim.x;
    __shared__ unsigned long long sb[NMS_BLK_C];

    for (int k = 0; k < MAX_DET_C; ++k) {
        unsigned long long key = __hip_atomic_load(&keys[k], __ATOMIC_ACQUIRE,
                                                   __HIP_MEMORY_SCOPE_AGENT);
        int pick = (key != 0ull) ? (int)(~(unsigned int)(key & 0xFFFFFFFFull))
                                 : -1;
        if (tid == 0) keep[k] = pick;

        unsigned long long best = 0ull;
        if (pick >= 0) {
            float4 pb = boxes[pick];
            float parea = (pb.z - pb.x) * (pb.w - pb.y);
            for (int i = tid; i < N_BOXES_C; i += total) {
                float s = scores[i];
                if (s < 0.0f) continue;               // already suppressed
                if (i == pick) { scores[i] = NEG_INF_F; continue; }
                float4 b = boxes[i];
                float xx1 = fmaxf(pb.x, b.x);
                float yy1 = fmaxf(pb.y, b.y);
                float xx2 = fminf(pb.z, b.z);
                float yy2 = fminf(pb.w, b.w);
                float iw  = fmaxf(xx2 - xx1, 0.0f);
                float ih  = fmaxf(yy2 - yy1, 0.0f);
                float inter = iw * ih;
                float area  = (b.z - b.x) * (b.w - b.y);
                float iou   = inter / ((parea + area) - inter); // ref op order
                if (iou > IOU_THR_C) {
                    scores[i] = NEG_INF_F;
                } else {
                    unsigned long long mk = make_key(s, (unsigned int)i);
                    if (mk > best) best = mk;
                }
            }
        }

        // block-level max of candidate keys for pick k+1
        sb[threadIdx.x] = best;
        __syncthreads();
        for (int off = NMS_BLK_C / 2; off > 0; off >>= 1) {
            if (threadIdx.x < off) {
                if (sb[threadIdx.x + off] > sb[threadIdx.x])
                    sb[threadIdx.x] = sb[threadIdx.x + off];
            }
            __syncthreads();
        }
        if (threadIdx.x == 0 && sb[0] != 0ull)
            atomicMax(&keys[k + 1], sb[0]);

        grid_barrier(&bar[0], &bar[1], gridDim.x);
    }
}

// ---------------------------------------------------------------------------
// Gather: out[k,:] = preds[keep[k],:] if keep[k] >= 0 else 0
// ---------------------------------------------------------------------------
__global__ void out_kernel(const float* __restrict__ preds,
                           const int* __restrict__ keep,
                           float* __restrict__ out) {
    int idx = blockIdx.x * blockDim.x + threadIdx.x;
    if (idx >= MAX_DET_C * N_FEATS_C) return;
    int k = idx / N_FEATS_C;
    int j = idx - k * N_FEATS_C;
    int kp = keep[k];
    out[idx] = (kp >= 0) ? preds[(size_t)kp * N_FEATS_C + j] : 0.0f;
}

extern "C" void kernel_launch(void* const* d_in, const int* in_sizes, int n_in,
                              void* d_out, int out_size, void* d_ws, size_t ws_size,
                              hipStream_t stream) {
    const float* preds = (const float*)d_in[0];   // (1, 100800, 85) fp32
    float* out = (float*)d_out;                   // (100, 85) fp32

    // Workspace layout (~2.02 MB, L2-resident):
    char* w = (char*)d_ws;
    float* boxes  = (float*)(w);                                   // N*4 f32
    float* scores = (float*)(w + (size_t)N_BOXES_C * 16);          // N f32
    unsigned long long* keys =
        (unsigned long long*)(w + (size_t)N_BOXES_C * 20);         // 101 u64
    int* keep = (int*)((char*)keys + (MAX_DET_C + 1) * 8);         // 100 i32
    unsigned int* bar = (unsigned int*)((char*)keep + MAX_DET_C * 4); // 2 u32

    nms_init_kernel<<<1, 128, 0, stream>>>(keys, bar);
    box_wmma_kernel<<<(NTILES_C + 7) / 8, 256, 0, stream>>>(preds, boxes);
    score_kernel<<<512, 256, 0, stream>>>(preds, scores, keys);
    nms_kernel<<<NMS_WG_C, NMS_BLK_C, 0, stream>>>(
        (const float4*)boxes, scores, keys, keep, bar);
    out_kernel<<<(MAX_DET_C * N_FEATS_C + 255) / 256, 256, 0, stream>>>(
        preds, keep, out);
}